// AttentionPooling_Idx_15960098472038
// MI455X (gfx1250) — compile-verified
//
#include <hip/hip_runtime.h>
#include <math.h>

// Problem constants (match reference)
#define BB 8
#define PP 16384
#define KK 16
#define DD 128
#define HH 64

typedef __attribute__((ext_vector_type(2))) float v2f;
typedef __attribute__((ext_vector_type(8))) float v8f;

// Branch-free fast transcendentals (v_exp_f32 / v_rcp_f32), ~1-2 ulp.
__device__ __forceinline__ float fast_tanh(float v) {
    const float a = fabsf(v);
    // 2*log2(e) = 2.8853900817779268
    const float e = __builtin_amdgcn_exp2f(a * 2.8853900817779268f);
    const float r = 1.0f - 2.0f * __builtin_amdgcn_rcpf(e + 1.0f); // tanh(|v|); e=inf -> 1
    return copysignf(r, v);
}

__device__ __forceinline__ float fast_sigmoid(float v) {
    // log2(e) = 1.4426950408889634
    const float e = __builtin_amdgcn_exp2f(-v * 1.4426950408889634f);
    return __builtin_amdgcn_rcpf(1.0f + e); // v << 0: e=inf -> 0
}

// ---------------------------------------------------------------------------
// Kernel 1: score MLP with fp32 WMMA.
// One wave per 16-point tile. Computes
//   xs[p] = sigmoid( sum_h V_w[h] * tanh( sum_d x[p,d]*W1_w[h,d] + W1_b[h] ) + V_b )
// A-matrix (16x4 f32): lane l<16 holds rows M=l, K=k0..k0+1; lanes 16..31 hold K=k0+2..k0+3.
// B-matrix (4x16 f32): lane l<16 holds cols N=l, K=k0..k0+1; lanes 16..31 hold K=k0+2..k0+3.
// C/D (16x16 f32, 8 VGPRs): vgpr r, lanes 0-15 -> (M=r, N=lane); lanes 16-31 -> (M=r+8, N=lane-16).
// ---------------------------------------------------------------------------
__global__ __launch_bounds__(256) void score_mlp_wmma(
    const float* __restrict__ x,      // (B*P, D)
    const float* __restrict__ W1_w,   // (H, D)
    const float* __restrict__ W1_b,   // (H)
    const float* __restrict__ V_w,    // (1, H)
    const float* __restrict__ V_b,    // (1)
    float* __restrict__ xs)           // (B*P)
{
    const int lane = threadIdx.x & 31;
    const int wave = threadIdx.x >> 5;
    const int tile = blockIdx.x * 8 + wave;       // 16-point tile id
    const size_t pt0 = (size_t)tile * 16;
    const int half = lane >> 4;                   // 0: K lo pair, 1: K hi pair
    const int l = lane & 15;

    const float* xrow = x + (pt0 + (size_t)l) * DD + half * 2;
    const float* w0 = W1_w + (size_t)( 0 + l) * DD + half * 2;
    const float* w1 = W1_w + (size_t)(16 + l) * DD + half * 2;
    const float* w2 = W1_w + (size_t)(32 + l) * DD + half * 2;
    const float* w3 = W1_w + (size_t)(48 + l) * DD + half * 2;

    v8f acc0 = {}, acc1 = {}, acc2 = {}, acc3 = {};

#pragma unroll 4
    for (int k0 = 0; k0 < DD; k0 += 4) {
        v2f a  = *(const v2f*)(xrow + k0);
        v2f b0 = *(const v2f*)(w0 + k0);
        v2f b1 = *(const v2f*)(w1 + k0);
        v2f b2 = *(const v2f*)(w2 + k0);
        v2f b3 = *(const v2f*)(w3 + k0);
        acc0 = __builtin_amdgcn_wmma_f32_16x16x4_f32(false, a, false, b0, (short)0, acc0, false, false);
        acc1 = __builtin_amdgcn_wmma_f32_16x16x4_f32(false, a, false, b1, (short)0, acc1, false, false);
        acc2 = __builtin_amdgcn_wmma_f32_16x16x4_f32(false, a, false, b2, (short)0, acc2, false, false);
        acc3 = __builtin_amdgcn_wmma_f32_16x16x4_f32(false, a, false, b3, (short)0, acc3, false, false);
    }

    // Apply bias + tanh, fold in V_w per column (this lane owns column h = n*16 + l
    // of accumulator n, for rows r+8*half).
    float part[8];
#pragma unroll
    for (int r = 0; r < 8; ++r) part[r] = 0.0f;

    auto fold = [&](const v8f& c, int n) {
        const float vw = V_w[n * 16 + l];
        const float bb = W1_b[n * 16 + l];
#pragma unroll
        for (int r = 0; r < 8; ++r)
            part[r] += vw * fast_tanh(c[r] + bb);
    };
    fold(acc0, 0); fold(acc1, 1); fold(acc2, 2); fold(acc3, 3);

    // Reduce across the 16 lanes of each half (xor of bits 0..3 stays inside a half).
    const float vb = V_b[0];
#pragma unroll
    for (int r = 0; r < 8; ++r) {
        float v = part[r];
        v += __shfl_xor(v, 1);
        v += __shfl_xor(v, 2);
        v += __shfl_xor(v, 4);
        v += __shfl_xor(v, 8);
        part[r] = fast_sigmoid(v + vb);
    }

    // Lane 0 writes rows 0..7 (half=0); lane 16 writes rows 8..15 (half=1).
    if (l == 0) {
#pragma unroll
        for (int r = 0; r < 8; ++r)
            xs[pt0 + (size_t)(half * 8 + r)] = part[r];
    }
}

// ---------------------------------------------------------------------------
// Kernel 2: neighbor gather + normalized weighted pooling.
// One wave per point. Lane l owns output floats 4l..4l+3 (b128 loads/stores).
// All gathered rows live in L2 (8 MB/batch slab << 192 MB L2).
// ---------------------------------------------------------------------------
__global__ __launch_bounds__(256) void attn_gather_pool(
    const float* __restrict__ x,      // (B*P, D)
    const int*   __restrict__ idx,    // (B*P, K)
    const float* __restrict__ xs,     // (B*P)
    float* __restrict__ out)          // (B*P, D)
{
    const int lane = threadIdx.x & 31;
    const int wave = threadIdx.x >> 5;
    const size_t point = (size_t)blockIdx.x * 8 + wave;   // 0 .. B*P-1
    const size_t b = point >> 14;                          // / P (P = 16384)

    const int*   ip  = idx + point * KK;
    const float* xb  = x  + (b << 14) * DD;                // batch base of x
    const float* xsb = xs + (b << 14);                     // batch base of scores

    int   myidx = 0;
    float s = 0.0f;
    if (lane < KK) {
        myidx = ip[lane];
        s = xsb[myidx];
    }

    // Sum of the 16 scores (lanes >= 16 contribute 0), broadcast from lane 0.
    float tot = s;
    tot += __shfl_xor(tot, 1);
    tot += __shfl_xor(tot, 2);
    tot += __shfl_xor(tot, 4);
    tot += __shfl_xor(tot, 8);
    tot = __shfl(tot, 0);
    const float inv = __builtin_amdgcn_rcpf(tot);

    float4 acc = make_float4(0.f, 0.f, 0.f, 0.f);
#pragma unroll
    for (int k = 0; k < KK; ++k) {
        const int   nb = __shfl(myidx, k);
        const float w  = __shfl(s, k) * inv;
        const float4 row = *(const float4*)(xb + (size_t)nb * DD + lane * 4);
        acc.x += w * row.x;
        acc.y += w * row.y;
        acc.z += w * row.z;
        acc.w += w * row.w;
    }
    *(float4*)(out + point * DD + lane * 4) = acc;
}

// ---------------------------------------------------------------------------
// Launch. Inputs (setup_inputs order): x, idx, W1_w, W1_b, V_w, V_b.
// d_out = [ out (B*P*D floats) | x_s (B*P floats) ]; we write x_s into the
// tail of d_out in kernel 1 and read it back (from L2) in kernel 2.
// ---------------------------------------------------------------------------
extern "C" void kernel_launch(void* const* d_in, const int* in_sizes, int n_in,
                              void* d_out, int out_size, void* d_ws, size_t ws_size,
                              hipStream_t stream) {
    const float* x    = (const float*)d_in[0];
    const int*   idx  = (const int*)  d_in[1];
    const float* W1_w = (const float*)d_in[2];
    const float* W1_b = (const float*)d_in[3];
    const float* V_w  = (const float*)d_in[4];
    const float* V_b  = (const float*)d_in[5];

    float* out = (float*)d_out;
    float* xs  = out + (size_t)BB * PP * DD;   // second tuple output

    // Kernel 1: (B*P)/16 = 8192 tiles, 8 waves (tiles) per 256-thread block.
    score_mlp_wmma<<<(BB * PP / 16) / 8, 256, 0, stream>>>(x, W1_w, W1_b, V_w, V_b, xs);

    // Kernel 2: B*P = 131072 points, 8 waves (points) per 256-thread block.
    attn_gather_pool<<<(BB * PP) / 8, 256, 0, stream>>>(x, idx, xs, out);
}